// ObliviousDecisionLayer_42915313221855
// MI455X (gfx1250) — compile-verified
//
#include <hip/hip_runtime.h>
#include <hip/hip_bf16.h>

typedef __attribute__((ext_vector_type(2))) float v2f;
typedef __attribute__((ext_vector_type(8))) float v8f;

#define NUM_TREES 256
#define DEPTH     6
#define BATCH     4096
#define FEATURES  256
#define NLEAF     64

// ---------------------------------------------------------------------------
// Row-sum of a (nrows x 256) f32 matrix via V_WMMA_F32_16X16X4_F32.
// One wave32 per 16-row tile:  D(16x16) = sum_k A(16x4) * Ones(4x16), so every
// column of D holds the exact f32 row sum. 64 K-chunks, 4 accumulators.
// ---------------------------------------------------------------------------
__global__ void rowsum_wmma_kernel(const float* __restrict__ src,
                                   float* __restrict__ dst, int nrows) {
    const int wave = (blockIdx.x * blockDim.x + threadIdx.x) >> 5;
    const int lane = threadIdx.x & 31;
    const int row0 = wave * 16;
    if (row0 >= nrows) return;   // wave-uniform; EXEC stays all-1s for WMMA

    const int m     = lane & 15;       // A-matrix M row for this lane
    const int khalf = lane >> 4;       // lanes 16-31 carry K=2,3 of each chunk
    const float* rowp = src + (size_t)(row0 + m) * FEATURES + khalf * 2;

    v2f ones; ones.x = 1.0f; ones.y = 1.0f;   // B = all-ones (layout-invariant)
    v8f c0 = {}, c1 = {}, c2 = {}, c3 = {};

#pragma unroll
    for (int kc = 0; kc < 16; ++kc) {
        v2f a0 = *(const v2f*)(rowp + (4 * kc + 0) * 4);
        v2f a1 = *(const v2f*)(rowp + (4 * kc + 1) * 4);
        v2f a2 = *(const v2f*)(rowp + (4 * kc + 2) * 4);
        v2f a3 = *(const v2f*)(rowp + (4 * kc + 3) * 4);
        c0 = __builtin_amdgcn_wmma_f32_16x16x4_f32(false, a0, false, ones,
                                                   (short)0, c0, false, false);
        c1 = __builtin_amdgcn_wmma_f32_16x16x4_f32(false, a1, false, ones,
                                                   (short)0, c1, false, false);
        c2 = __builtin_amdgcn_wmma_f32_16x16x4_f32(false, a2, false, ones,
                                                   (short)0, c2, false, false);
        c3 = __builtin_amdgcn_wmma_f32_16x16x4_f32(false, a3, false, ones,
                                                   (short)0, c3, false, false);
    }
    v8f c = c0 + c1 + c2 + c3;   // disjoint K-chunks: sums add exactly

    // D layout: lane 0 vgpr r -> (M=r, N=0); lane 16 vgpr r -> (M=8+r, N=0)
    if (lane == 0) {
#pragma unroll
        for (int r = 0; r < 8; ++r) dst[row0 + r] = c[r];
    } else if (lane == 16) {
#pragma unroll
        for (int r = 0; r < 8; ++r) dst[row0 + 8 + r] = c[r];
    }
}

// ---------------------------------------------------------------------------
// One block per batch element b; thread t evaluates tree t and the block
// reduces the mean over trees.  decisions[b,t,d] = sigmoid(s[b]*w[t,d]+thr),
// leaf fold: bit d of leaf index -> sigma_d (bit=0) vs 1-sigma_d (bit=1).
// ---------------------------------------------------------------------------
__global__ void forest_eval_kernel(const float* __restrict__ s,
                                   const float* __restrict__ wsum,
                                   const float* __restrict__ thr,
                                   const float* __restrict__ leaf,
                                   float* __restrict__ out) {
    const int b = blockIdx.x;
    const int t = threadIdx.x;          // 256 trees
    const float sb = s[b];

    float sig[DEPTH];
#pragma unroll
    for (int d = 0; d < DEPTH; ++d) {
        const float z = sb * wsum[t * DEPTH + d] + thr[t * DEPTH + d];
        sig[d] = 1.0f / (1.0f + __expf(-z));
    }

    // Fold 64 leaves down, consuming bit 0 (depth 0) first.
    const float* lp = leaf + t * NLEAF;
    float v[32];
#pragma unroll
    for (int j = 0; j < 32; ++j)
        v[j] = sig[0] * lp[2 * j] + (1.0f - sig[0]) * lp[2 * j + 1];
#pragma unroll
    for (int d = 1; d < DEPTH; ++d) {
        const int len = 32 >> d;
#pragma unroll
        for (int j = 0; j < len; ++j)
            v[j] = sig[d] * v[2 * j] + (1.0f - sig[d]) * v[2 * j + 1];
    }

    __shared__ float red[NUM_TREES];
    red[t] = v[0];
    __syncthreads();
#pragma unroll
    for (int off = NUM_TREES / 2; off > 0; off >>= 1) {
        if (t < off) red[t] += red[t + off];
        __syncthreads();
    }
    if (t == 0) out[b] = red[0] * (1.0f / (float)NUM_TREES);
}

extern "C" void kernel_launch(void* const* d_in, const int* in_sizes, int n_in,
                              void* d_out, int out_size, void* d_ws, size_t ws_size,
                              hipStream_t stream) {
    const float* inputs = (const float*)d_in[0];   // (4096, 256)
    const float* dw     = (const float*)d_in[1];   // (256, 6, 256)
    const float* thr    = (const float*)d_in[2];   // (256, 6)
    const float* leaf   = (const float*)d_in[3];   // (256, 64)
    float* out = (float*)d_out;                    // (4096, 1)

    float* s    = (float*)d_ws;                    // 4096 row sums of inputs
    float* wsum = s + BATCH;                       // 1536 row sums of weights

    // 8 waves per 256-thread block, one wave per 16-row tile.
    rowsum_wmma_kernel<<<(BATCH / 16) / 8, 256, 0, stream>>>(inputs, s, BATCH);
    rowsum_wmma_kernel<<<(NUM_TREES * DEPTH / 16) / 8, 256, 0, stream>>>(
        dw, wsum, NUM_TREES * DEPTH);
    forest_eval_kernel<<<BATCH, NUM_TREES, 0, stream>>>(s, wsum, thr, leaf, out);
}